// ChamferDistance_6468220748249
// MI455X (gfx1250) — compile-verified
//
#include <hip/hip_runtime.h>

typedef __attribute__((ext_vector_type(2))) float v2f;
typedef __attribute__((ext_vector_type(8))) float v8f;

#define WAVES   8
#define BLOCK   (WAVES * 32)     // 256 threads = 8 wave32
#define QPW     16               // query points per wave (WMMA M dim)
#define QPB     (WAVES * QPW)    // 128 query points per block
#define TCHUNK  1024             // staged target points per block iteration
#define HPAD    16               // 128B pad between half views (bank-group split)

// One direction of chamfer: for each query point, min squared distance (and argmin
// index) over all target points. Cross terms via V_WMMA_F32_16X16X4_F32 (K=3 padded
// to 4). Query norm is row-constant so it is folded OUT of the inner-loop argmin
// (argmin_n(qn+tn-2dot) == argmin_n(tn-2dot)) and added back once at the end.
__global__ __launch_bounds__(BLOCK)
void chamfer_dir_kernel(const float* __restrict__ qpts,   // [B, Nq, 3]
                        const float* __restrict__ tpts,   // [B, Mt, 3]
                        float* __restrict__ odist,        // [B, Nq]
                        int*   __restrict__ oidx,         // [B, Nq]
                        int Nq, int Mt)
{
    // Half-split B-matrix views: [0..TCHUNK) = (x,y) for lanes 0-15,
    // [TCHUNK+HPAD..) = (z,0) for lanes 16-31.  HPAD (=128B) keeps the two
    // views on disjoint LDS bank groups for conflict-free ds_load_b64.
    __shared__ float2 sPair[2 * TCHUNK + HPAD];
    __shared__ float  sN[TCHUNK];          // |t|^2

    const int tid  = threadIdx.x;
    const int lane = tid & 31;
    const int wave = tid >> 5;
    const int half = lane >> 4;
    const int ln   = lane & 15;

    // per-batch base pointers
    qpts  += (size_t)blockIdx.y * Nq * 3;
    tpts  += (size_t)blockIdx.y * Mt * 3;
    odist += (size_t)blockIdx.y * Nq;
    oidx  += (size_t)blockIdx.y * Nq;

    const int qbase = (blockIdx.x * WAVES + wave) * QPW;

    // ---- Load this wave's 16 query points (lane ln holds query qbase+ln in both halves)
    int qm  = qbase + ln;
    int qmc = qm < Nq ? qm : (Nq - 1);          // clamp: keep EXEC full for WMMA
    float qx = qpts[3 * qmc + 0];
    float qy = qpts[3 * qmc + 1];
    float qz = qpts[3 * qmc + 2];
    float qnorm = qx * qx + qy * qy + qz * qz;

    // A-matrix (16x4, MxK): lane l, VGPR r holds A[l%16][r + 2*(l/16)]
    // coords padded to (x, y, z, 0): half0 -> (x,y), half1 -> (z,0)
    v2f amat;
    amat.x = half ? qz : qx;
    amat.y = half ? 0.0f : qy;

    // Accumulator row v of the 16x16 D tile corresponds to query m = v + 8*half.
    // Broadcast that query's norm to every lane (ds_bpermute); used only at the end.
    float qn[8];
#pragma unroll
    for (int v = 0; v < 8; ++v)
        qn[v] = __shfl(qnorm, v + (half << 3), 32);

    float best[8];                 // running min of (tn - 2*dot) per row
    int   bidx[8];
#pragma unroll
    for (int v = 0; v < 8; ++v) { best[v] = __builtin_inff(); bidx[v] = 0; }

    const int pbase = half ? (TCHUNK + HPAD) : 0;   // which half-view this lane reads

    for (int tbase = 0; tbase < Mt; tbase += TCHUNK) {
        __syncthreads();   // previous chunk's consumers done before restage
        // ---- Cooperative stage: split-view pairs + norms
        for (int j = tid; j < TCHUNK; j += BLOCK) {
            int g = tbase + j;
            float x = 0.0f, y = 0.0f, z = 0.0f, n = __builtin_inff();
            if (g < Mt) {
                x = tpts[3 * g + 0];
                y = tpts[3 * g + 1];
                z = tpts[3 * g + 2];
                n = x * x + y * y + z * z;
            }
            sPair[j]                 = make_float2(x, y);   // lanes 0-15 view (K=0,1)
            sPair[TCHUNK + HPAD + j] = make_float2(z, 0.f); // lanes 16-31 view (K=2,3)
            sN[j] = n;
        }
        __syncthreads();

        // ---- 16 targets per WMMA tile
#pragma unroll 2
        for (int sub = 0; sub < TCHUNK / 16; ++sub) {
            float2 bp = sPair[pbase + sub * 16 + ln];   // ds_load_b64, no cndmask
            float  tn = sN[sub * 16 + ln];              // ds_load_b32 (same-addr bcast per pair)
            v2f bmat;
            bmat.x = bp.x;
            bmat.y = bp.y;

            v8f c = {};
            // acc[v] @ lane l = dot(query[v + 8*(l/16)], target[l%16])
            v8f acc = __builtin_amdgcn_wmma_f32_16x16x4_f32(
                false, amat, false, bmat, (short)0, c, false, false);

            int nidx = tbase + sub * 16 + ln;
#pragma unroll
            for (int v = 0; v < 8; ++v) {
                float d = __builtin_fmaf(-2.0f, acc[v], tn);   // qn folded out
                if (d < best[v]) { best[v] = d; bidx[v] = nidx; }  // strict <: first index wins
            }
        }
    }

    // ---- Final argmin reduction across the 16 lanes of each half (xor stays in-half)
#pragma unroll
    for (int v = 0; v < 8; ++v) {
        float b  = best[v];
        int   bi = bidx[v];
#pragma unroll
        for (int off = 8; off >= 1; off >>= 1) {
            float ob  = __shfl_xor(b, off, 32);
            int   obi = __shfl_xor(bi, off, 32);
            if (ob < b || (ob == b && obi < bi)) { b = ob; bi = obi; }
        }
        if (ln == 0) {
            int m = qbase + v + (half << 3);
            if (m < Nq) { odist[m] = qn[v] + b; oidx[m] = bi; }   // add row norm back here
        }
    }
}

extern "C" void kernel_launch(void* const* d_in, const int* in_sizes, int n_in,
                              void* d_out, int out_size, void* d_ws, size_t ws_size,
                              hipStream_t stream) {
    (void)n_in; (void)out_size; (void)d_ws; (void)ws_size;

    const float* xyz1 = (const float*)d_in[0];   // [B, N, 3]
    const float* xyz2 = (const float*)d_in[1];   // [B, M, 3]

    const int B = 2;
    const int N = in_sizes[0] / (3 * B);
    const int M = in_sizes[1] / (3 * B);

    // Output: dist1[B*N] f32 | dist2[B*M] f32 | idx1[B*N] i32 | idx2[B*M] i32
    float* out   = (float*)d_out;
    float* dist1 = out;
    float* dist2 = out + (size_t)B * N;
    int*   idx1  = (int*)(out + (size_t)B * (N + M));
    int*   idx2  = idx1 + (size_t)B * N;

    dim3 blk(BLOCK);
    dim3 g1((N + QPB - 1) / QPB, B);
    dim3 g2((M + QPB - 1) / QPB, B);

    // direction 1: for each point in xyz1, nearest in xyz2
    chamfer_dir_kernel<<<g1, blk, 0, stream>>>(xyz1, xyz2, dist1, idx1, N, M);
    // direction 2: for each point in xyz2, nearest in xyz1
    chamfer_dir_kernel<<<g2, blk, 0, stream>>>(xyz2, xyz1, dist2, idx2, M, N);
}